// ConvSSMBlockFourier_13091060318318
// MI455X (gfx1250) — compile-verified
//
#include <hip/hip_runtime.h>

// ---------------------------------------------------------------------------
// ConvSSMBlockFourier on MI455X (gfx1250):
//   effective spatial kernel G_c = sum_{j=0}^{7} B_c (*) A_dec_c^{(*)j}  (49x49)
//   y = depthwise linear conv(x, G)   via Toeplitz-tiled V_WMMA_F32_16X16X4_F32
// ---------------------------------------------------------------------------

typedef float v2f __attribute__((ext_vector_type(2)));
typedef float v8f __attribute__((ext_vector_type(8)));

#define CCH      256
#define BATCH    4
#define HH       128
#define WW       128
#define PLANE    (HH * WW)
#define KK       7
#define GS       49          // effective kernel support
#define GSTRIDE  64          // padded row stride of G in workspace
#define TSTEPS   8
#define DECAYR   0.3f

// workspace layout (in floats)
#define G_ELEMS  (CCH * GS * GSTRIDE)            // 802,816
#define XT_OFF   (G_ELEMS)
#define XT_ELEMS (CCH * BATCH * PLANE)           // 16,777,216
#define YT_OFF   (XT_OFF + XT_ELEMS)

// ---------------------------------------------------------------------------
// Kernel 1: build per-channel effective kernel G_c (49x49) in LDS.
//   cur_0 = B ; G = sum_j cur_j ; cur_{j+1} = cur_j (*) A_dec
// ---------------------------------------------------------------------------
__global__ __launch_bounds__(256) void build_g_kernel(
    const float* __restrict__ Ak, const float* __restrict__ Bk,
    float* __restrict__ g) {
  __shared__ float Ad[KK * KK];
  __shared__ float cur[GS * 52];
  __shared__ float nxt[GS * 52];
  __shared__ float gac[GS * 52];

  const int c = blockIdx.x;
  const int t = threadIdx.x;

  if (t < KK * KK) {
    int p = t / KK, q = t % KK;
    Ad[t] = Ak[c * KK * KK + t] * __expf(-DECAYR * (float)p) *
            __expf(-DECAYR * (float)q);
  }
  for (int idx = t; idx < GS * 52; idx += 256) { cur[idx] = 0.f; gac[idx] = 0.f; }
  __syncthreads();
  if (t < KK * KK) {
    int p = t / KK, q = t % KK;
    float bv = Bk[c * KK * KK + t];
    cur[p * 52 + q] = bv;
    gac[p * 52 + q] = bv;
  }
  __syncthreads();

  for (int j = 1; j < TSTEPS; ++j) {
    for (int idx = t; idx < GS * GS; idx += 256) {
      int r = idx / GS, s = idx % GS;
      float acc = 0.f;
      #pragma unroll
      for (int p = 0; p < KK; ++p) {
        int rr = r - p;
        if (rr >= 0) {
          #pragma unroll
          for (int q = 0; q < KK; ++q) {
            int ss = s - q;
            if (ss >= 0) acc += cur[rr * 52 + ss] * Ad[p * KK + q];
          }
        }
      }
      nxt[r * 52 + s] = acc;
    }
    __syncthreads();
    for (int idx = t; idx < GS * GS; idx += 256) {
      int r = idx / GS, s = idx % GS;
      float v = nxt[r * 52 + s];
      cur[r * 52 + s] = v;
      gac[r * 52 + s] += v;
    }
    __syncthreads();
  }

  for (int idx = t; idx < GS * GSTRIDE; idx += 256) {
    int r = idx >> 6, q = idx & 63;
    g[c * GS * GSTRIDE + idx] = (q < GS) ? gac[r * 52 + q] : 0.f;
  }
}

// ---------------------------------------------------------------------------
// Kernel 2: transpose x (B,H,W,C) -> per-channel planes xt[(c*B+b)*H*W + i*W + j]
// ---------------------------------------------------------------------------
__global__ __launch_bounds__(256) void transpose_fwd_kernel(
    const float* __restrict__ x, float* __restrict__ xt) {
  __shared__ float tile[32][33];
  unsigned id = blockIdx.x;
  int cb = id & 7;  id >>= 3;
  int jb = id & 3;  id >>= 2;
  int i  = id & 127; id >>= 7;
  int b  = id;
  const int t  = threadIdx.x;
  const int l0 = t >> 5;   // 0..7
  const int l1 = t & 31;   // 0..31
  #pragma unroll
  for (int r = 0; r < 4; ++r) {
    int jj = r * 8 + l0;
    tile[jj][l1] = x[(((b * HH + i) * WW + jb * 32 + jj) * CCH) + cb * 32 + l1];
  }
  __syncthreads();
  #pragma unroll
  for (int r = 0; r < 4; ++r) {
    int cc = r * 8 + l0;
    xt[(((cb * 32 + cc) * BATCH + b) * PLANE) + i * WW + jb * 32 + l1] =
        tile[l1][cc];
  }
}

// ---------------------------------------------------------------------------
// Kernel 3: depthwise conv via banded-Toeplitz WMMA (f32 16x16x4).
// Block = one (channel,batch) plane x one 32-row group.  8 waves; wave w owns
// column-tile tj=w and two row-tiles (rg*32, rg*32+16).
// LDS: 80-row input slab (48-row zero halo) + zero-padded G column table
// (16 leading + 15 trailing zeros per column -> no range selects on A loads).
// ---------------------------------------------------------------------------
#define XSTRIDE   136        // 2*136 % 64 == 16 -> lane-halves hit disjoint banks
#define SLAB_ROWS 80
#define GCOL      80         // padded G column length: [16 zeros | 49 vals | 15 zeros]

__global__ __launch_bounds__(256) void conv_wmma_kernel(
    const float* __restrict__ xt, const float* __restrict__ g,
    float* __restrict__ yt) {
  __shared__ float xpl[SLAB_ROWS * XSTRIDE];   // 43,520 B
  __shared__ float gT[GS * GCOL];              // 15,680 B : gT[q*80 + 16 + r]

  const int rg    = blockIdx.x & 3;
  const int plane = blockIdx.x >> 2;           // plane = c*BATCH + b
  const int c     = plane >> 2;
  const int t     = threadIdx.x;

  // phase 0: zero the padded G column table
  for (int idx = t; idx < GS * GCOL; idx += 256) gT[idx] = 0.f;
  __syncthreads();

  // phase 1: scatter G (row-major r x 64) -> gT[q][16 + r], and load the slab
  const float* gc = g + c * GS * GSTRIDE;
  for (int idx = t; idx < GS * GSTRIDE; idx += 256) {
    int r = idx >> 6, q = idx & 63;
    float v = gc[idx];
    if (q < GS) gT[q * GCOL + 16 + r] = v;
  }
  const int row0 = rg * 32 - 48;
  const float* xp = xt + plane * PLANE;
  for (int idx = t; idx < SLAB_ROWS * WW; idx += 256) {
    int rr = idx >> 7, ccx = idx & 127;
    int gr = row0 + rr;
    xpl[rr * XSTRIDE + ccx] = (gr >= 0) ? xp[gr * WW + ccx] : 0.f;
  }
  __syncthreads();

  const int lane = t & 31;
  const int w    = t >> 5;        // wave id == column tile tj
  const int half = lane >> 4;     // lane half: selects K pair / M half
  const int m    = lane & 15;     // A row (M) and B col (N) for this lane

  v8f acc0 = {0.f, 0.f, 0.f, 0.f, 0.f, 0.f, 0.f, 0.f};
  v8f acc1 = {0.f, 0.f, 0.f, 0.f, 0.f, 0.f, 0.f, 0.f};

  const int rbase = m + 48 - 2 * half;   // A row for kb=0, K=2h
  const int brow  = 2 * half;            // B slab row offset for K=2h

  if (w >= 3) {
    // fast path: colv = 16w - q + m >= 0 always -> no B-range selects
    for (int q = 0; q < GS; ++q) {
      const float* gq = gT + q * GCOL + 16;
      const float* xr = xpl + (w * 16 - q + m);
      #pragma unroll
      for (int kb = 0; kb < 16; ++kb) {
        int rx = rbase - 4 * kb;
        v2f a  = { gq[rx], gq[rx - 1] };
        int br = 4 * kb + brow;
        v2f b0 = { xr[br * XSTRIDE],        xr[(br + 1) * XSTRIDE] };
        v2f b1 = { xr[(br + 16) * XSTRIDE], xr[(br + 17) * XSTRIDE] };
        acc0 = __builtin_amdgcn_wmma_f32_16x16x4_f32(
            false, a, false, b0, (short)0, acc0, false, false);
        acc1 = __builtin_amdgcn_wmma_f32_16x16x4_f32(
            false, a, false, b1, (short)0, acc1, false, false);
      }
    }
  } else {
    // left-edge path: column index may be negative -> clamp + select
    for (int q = 0; q < GS; ++q) {
      const int colv = w * 16 - q + m;
      const bool cok = colv >= 0;
      const float* gq = gT + q * GCOL + 16;
      const float* xr = xpl + (cok ? colv : 0);
      #pragma unroll
      for (int kb = 0; kb < 16; ++kb) {
        int rx = rbase - 4 * kb;
        v2f a  = { gq[rx], gq[rx - 1] };
        int br = 4 * kb + brow;
        float t00 = xr[br * XSTRIDE];
        float t01 = xr[(br + 1) * XSTRIDE];
        float t10 = xr[(br + 16) * XSTRIDE];
        float t11 = xr[(br + 17) * XSTRIDE];
        v2f b0, b1;
        b0.x = cok ? t00 : 0.f;  b0.y = cok ? t01 : 0.f;
        b1.x = cok ? t10 : 0.f;  b1.y = cok ? t11 : 0.f;
        acc0 = __builtin_amdgcn_wmma_f32_16x16x4_f32(
            false, a, false, b0, (short)0, acc0, false, false);
        acc1 = __builtin_amdgcn_wmma_f32_16x16x4_f32(
            false, a, false, b1, (short)0, acc1, false, false);
      }
    }
  }

  // D layout: vgpr v, lane-half h, lane n -> element (M = v + 8h, N = n)
  float* yp = yt + plane * PLANE;
  const int i0a = rg * 32, i0b = i0a + 16;
  const int j0  = w * 16;
  #pragma unroll
  for (int v = 0; v < 8; ++v) {
    yp[(i0a + v + 8 * half) * WW + j0 + m] = acc0[v];
    yp[(i0b + v + 8 * half) * WW + j0 + m] = acc1[v];
  }
}

// ---------------------------------------------------------------------------
// Kernel 4: transpose yt (C,B,H,W) back to channel-last output (B,H,W,C)
// ---------------------------------------------------------------------------
__global__ __launch_bounds__(256) void transpose_bwd_kernel(
    const float* __restrict__ yt, float* __restrict__ y) {
  __shared__ float tile[32][33];
  unsigned id = blockIdx.x;
  int cb = id & 7;  id >>= 3;
  int jb = id & 3;  id >>= 2;
  int i  = id & 127; id >>= 7;
  int b  = id;
  const int t  = threadIdx.x;
  const int l0 = t >> 5;
  const int l1 = t & 31;
  #pragma unroll
  for (int r = 0; r < 4; ++r) {
    int cc = r * 8 + l0;
    tile[cc][l1] = yt[((cb * 32 + cc) * BATCH + b) * PLANE + i * WW + jb * 32 + l1];
  }
  __syncthreads();
  #pragma unroll
  for (int r = 0; r < 4; ++r) {
    int jj = r * 8 + l0;
    y[(((b * HH + i) * WW + jb * 32 + jj) * CCH) + cb * 32 + l1] = tile[l1][jj];
  }
}

// ---------------------------------------------------------------------------
extern "C" void kernel_launch(void* const* d_in, const int* in_sizes, int n_in,
                              void* d_out, int out_size, void* d_ws, size_t ws_size,
                              hipStream_t stream) {
  const float* x  = (const float*)d_in[0];
  const float* Ak = (const float*)d_in[1];
  const float* Bk = (const float*)d_in[2];
  float* ws = (float*)d_ws;
  float* g  = ws;
  float* xt = ws + XT_OFF;
  float* yt = ws + YT_OFF;
  float* y  = (float*)d_out;

  build_g_kernel<<<CCH, 256, 0, stream>>>(Ak, Bk, g);
  transpose_fwd_kernel<<<BATCH * HH * 4 * 8, 256, 0, stream>>>(x, xt);
  conv_wmma_kernel<<<CCH * BATCH * 4, 256, 0, stream>>>(xt, g, yt);
  transpose_bwd_kernel<<<BATCH * HH * 4 * 8, 256, 0, stream>>>(yt, y);
}